// CharacterClustering_52278341926957
// MI455X (gfx1250) — compile-verified
//
#include <hip/hip_runtime.h>

// Problem constants (match reference)
#define BB 16
#define SS 8192
#define DD 512
#define WW (SS / 2 + 1)          // 4097 word slots
#define NGROUP ((WW + 15) / 16)  // 257 groups of 16 words

typedef __attribute__((ext_vector_type(2))) float v2f;
typedef __attribute__((ext_vector_type(8))) float v8f;

__device__ __forceinline__ bool is_boundary(int c) {
    return c == 32 || c == 44 || c == 46 || c == 33 ||
           c == 63 || c == 10 || c == 13 || c == 9;
}

// ---------------------------------------------------------------------------
// Kernel 1: per-batch word segmentation. One 1024-thread block per batch,
// 8 chars per thread, block-wide inclusive scan of word-starts.
// Produces seg[B,S], word_start[B,W], word_end[B,W], num_words[B].
// ---------------------------------------------------------------------------
__global__ __launch_bounds__(1024) void seg_scan_kernel(
    const int* __restrict__ char_ids,
    int* __restrict__ seg,
    int* __restrict__ wstart,
    int* __restrict__ wend,
    int* __restrict__ nww,
    float* __restrict__ out_nw)
{
    const int b   = blockIdx.x;
    const int tid = threadIdx.x;
    const int s0  = tid * 8;
    const int* ids = char_ids + (size_t)b * SS;

    int  cid[8];
    bool bnd[8];
#pragma unroll
    for (int j = 0; j < 8; ++j) cid[j] = ids[s0 + j];
#pragma unroll
    for (int j = 0; j < 8; ++j) bnd[j] = is_boundary(cid[j]);

    const bool prev0 = (s0 == 0)      ? true : is_boundary(ids[s0 - 1]);
    const bool next7 = (s0 + 8 >= SS) ? true : is_boundary(ids[s0 + 8]);

    int starts[8], ls[8];
    int run = 0;
#pragma unroll
    for (int j = 0; j < 8; ++j) {
        const bool pb = (j == 0) ? prev0 : bnd[j - 1];
        starts[j] = (!bnd[j] && pb) ? 1 : 0;
        run += starts[j];
        ls[j] = run;
    }

    __shared__ int sc[1024];
    sc[tid] = run;
    __syncthreads();
    for (int off = 1; off < 1024; off <<= 1) {
        const int v = (tid >= off) ? sc[tid - off] : 0;
        __syncthreads();
        sc[tid] += v;
        __syncthreads();
    }
    const int excl = sc[tid] - run;   // exclusive prefix of word-starts
    const int nwv  = sc[1023];        // total words in this batch

#pragma unroll
    for (int j = 0; j < 8; ++j) {
        const int s = s0 + j;
        int sv;
        if (!bnd[j]) {
            sv = excl + ls[j] - 1;                       // segment id
            if (starts[j]) wstart[(size_t)b * WW + sv] = s;
            const bool nb = (j == 7) ? next7 : bnd[j + 1];
            if (nb) wend[(size_t)b * WW + sv] = s;       // last char of run
        } else {
            sv = WW;                                     // trash segment
        }
        seg[(size_t)b * SS + s] = sv;
    }
    if (tid == 0) {
        nww[b]    = nwv;
        out_nw[b] = (float)nwv;
    }
}

// ---------------------------------------------------------------------------
// Kernel 2: WMMA segmented mean-pool. Grid = (257 word-groups, 16 batches),
// 128 threads = 4 waves. Group g owns words [16g, 16g+16): a contiguous char
// range [cs, ce]. Each wave owns 8 contiguous 16-dim tiles (128 dims) with 8
// live f32 accumulators; the K-loop is branch-free (LDS seg padded with the
// trash id, embedding addresses clamped to ce) and issues 16 global loads +
// 8 v_wmma_f32_16x16x4_f32 per 4-char step.
// ---------------------------------------------------------------------------
__global__ __launch_bounds__(128) void pool_wmma_kernel(
    const float* __restrict__ emb,
    const int* __restrict__ seg,
    const int* __restrict__ wstart,
    const int* __restrict__ wend,
    const int* __restrict__ nww,
    float* __restrict__ out_vecs,
    float* __restrict__ out_counts)
{
    const int b   = blockIdx.y;
    const int g   = blockIdx.x;
    const int w0  = g * 16;
    const int tid = threadIdx.x;
    const int nw  = nww[b];

    __shared__ float s_rinv[16];
    __shared__ int   s_cs, s_ce;
    __shared__ int   s_seg[SS + 4];

    if (tid < 16) {
        const int w = w0 + tid;
        int cnt = 0;
        if (w < nw) cnt = wend[(size_t)b * WW + w] - wstart[(size_t)b * WW + w] + 1;
        s_rinv[tid] = (cnt > 0) ? (1.0f / (float)cnt) : 0.0f;
        if (w < WW) out_counts[(size_t)b * WW + w] = (float)cnt;
    }
    if (tid == 0) {
        if (w0 < nw) {
            s_cs = wstart[(size_t)b * WW + w0];
            const int wl = min(w0 + 15, nw - 1);
            s_ce = wend[(size_t)b * WW + wl];
        } else {
            s_cs = 1; s_ce = 0;                          // empty group
        }
    }
    __syncthreads();

    const int cs = s_cs, ce = s_ce;

    if (cs > ce) {
        // No valid words in this group: vectorized zero-fill of its rows.
        const int rows = min(16, WW - w0);
        if (rows > 0) {
            float4* dst = (float4*)(out_vecs + ((size_t)b * WW + w0) * DD);
            const int n4 = rows * (DD / 4);
            for (int i = tid; i < n4; i += 128)
                dst[i] = make_float4(0.f, 0.f, 0.f, 0.f);
        }
        return;
    }

    // Stage this group's segment ids in LDS, padded to a multiple of 4 with
    // the trash id WW so the K-loop needs no tail guards.
    const int nchars = ce - cs + 1;
    const int np     = (nchars + 3) & ~3;
    for (int i = tid; i < np; i += 128)
        s_seg[i] = (i < nchars) ? seg[(size_t)b * SS + cs + i] : WW;
    __syncthreads();

    const int wave = tid >> 5;
    const int lane = tid & 31;
    const int half = lane >> 4;       // K half: lanes 0-15 -> K{0,1}, 16-31 -> K{2,3}
    const int col  = lane & 15;       // A row (word) / B col (dim)
    // Fold the targ < WW guard into a sentinel: seg values are >= 0, so -1
    // never matches (excludes the trash segment and out-of-range word rows).
    const int targ = (w0 + col < WW) ? (w0 + col) : -1;

    // This wave's 128 contiguous dims: tiles tbase..tbase+7.
    const int tbase = wave * 8;
    const float* ecol = emb + (size_t)b * SS * DD + tbase * 16 + col;

    v8f acc[8];
#pragma unroll
    for (int j = 0; j < 8; ++j) acc[j] = (v8f){0.f,0.f,0.f,0.f,0.f,0.f,0.f,0.f};

    for (int kk = 0; kk < np; kk += 4) {
        const int i0 = kk + half * 2;          // this lane's two K slots
        const int i1 = i0 + 1;
        const int sv0 = s_seg[i0];
        const int sv1 = s_seg[i1];
        const float a0 = (sv0 == targ) ? 1.0f : 0.0f;
        const float a1 = (sv1 == targ) ? 1.0f : 0.0f;
        const v2f Av = {a0, a1};
        // Clamp to ce: padded slots read valid (finite) data, masked by a=0.
        const int c0 = min(cs + i0, ce);
        const int c1 = min(cs + i1, ce);
        const float* row0 = ecol + (size_t)c0 * DD;
        const float* row1 = ecol + (size_t)c1 * DD;
#pragma unroll
        for (int j = 0; j < 8; ++j) {
            const v2f Bv = {row0[j * 16], row1[j * 16]};
            acc[j] = __builtin_amdgcn_wmma_f32_16x16x4_f32(
                false, Av, false, Bv, (short)0, acc[j], false, false);
        }
    }

    // Preload the per-row scales this lane needs (rows half*8 .. half*8+7).
    float rv[8];
#pragma unroll
    for (int r = 0; r < 8; ++r) rv[r] = s_rinv[r + half * 8];

    float* obase = out_vecs + ((size_t)b * WW + w0 + half * 8) * DD + tbase * 16 + col;

    if (w0 + 16 <= WW) {
        // Fast path (256 of 257 groups): no row guards.
#pragma unroll
        for (int r = 0; r < 8; ++r)
#pragma unroll
            for (int j = 0; j < 8; ++j)
                obase[(size_t)r * DD + j * 16] = acc[j][r] * rv[r];
    } else {
#pragma unroll
        for (int r = 0; r < 8; ++r) {
            if (w0 + r + half * 8 < WW) {
#pragma unroll
                for (int j = 0; j < 8; ++j)
                    obase[(size_t)r * DD + j * 16] = acc[j][r] * rv[r];
            }
        }
    }
}

// ---------------------------------------------------------------------------
extern "C" void kernel_launch(void* const* d_in, const int* in_sizes, int n_in,
                              void* d_out, int out_size, void* d_ws, size_t ws_size,
                              hipStream_t stream) {
    const float* emb      = (const float*)d_in[0];   // [B,S,D] f32
    const int*   char_ids = (const int*)d_in[1];     // [B,S] i32

    float* out        = (float*)d_out;
    float* out_vecs   = out;                                   // [B,W,D]
    float* out_counts = out + (size_t)BB * WW * DD;            // [B,W]
    float* out_nw     = out_counts + (size_t)BB * WW;          // [B]

    int* seg = (int*)d_ws;                                     // [B,S]
    int* wst = seg + (size_t)BB * SS;                          // [B,W]
    int* wen = wst + (size_t)BB * WW;                          // [B,W]
    int* nww = wen + (size_t)BB * WW;                          // [B]

    hipLaunchKernelGGL(seg_scan_kernel, dim3(BB), dim3(1024), 0, stream,
                       char_ids, seg, wst, wen, nww, out_nw);
    hipLaunchKernelGGL(pool_wmma_kernel, dim3(NGROUP, BB), dim3(128), 0, stream,
                       emb, seg, wst, wen, nww, out_vecs, out_counts);
}